// QMLP_19567871000852
// MI455X (gfx1250) — compile-verified
//
#include <hip/hip_runtime.h>
#include <hip/hip_bf16.h>
#include <math.h>

// ---------------------------------------------------------------------------
// QMLP: pool+compress -> 16-qubit statevector circuit (2 layers) -> <Z_i> -> FC
// State: 128 x 65536 complex64 = 64 MB, resident in MI455X 192MB L2.
// Gate passes fused into 16x16 group unitaries applied as realified 32x32
// matmuls with V_WMMA_F32_16X16X4_F32 (wave32). 22 in-place passes total.
// Workspace requirement: ~72 MB.
// Qubit w of the reference maps to bit (15-w) of the flat state index.
// ---------------------------------------------------------------------------

typedef float v2f __attribute__((ext_vector_type(2)));
typedef float v8f __attribute__((ext_vector_type(8)));

#define B_SZ 128

// ---------------- Stage 0: adaptive pool (4,4,1) ---------------------------
// x: (128, 2, 16, 64, 64) f32. One block per output element (b, j), j = c*16+dblk*4+hblk.
__global__ void pool_kernel(const float* __restrict__ x, float* __restrict__ p) {
  int bj = blockIdx.x;            // 0..4095
  int b = bj >> 5, j = bj & 31;
  int c = j >> 4, dblk = (j >> 2) & 3, hblk = j & 3;
  const float* base = x + ((((size_t)(b * 2 + c) * 16 + dblk * 4) * 64 + hblk * 16) * 64);
  float s = 0.f;
  for (int u = 0; u < 16; ++u) {
    int e = threadIdx.x * 16 + u;           // 0..4095 over (4 d, 16 h, 64 w)
    int d_off = e >> 10, h_off = (e >> 6) & 15, w = e & 63;
    s += base[((size_t)d_off * 64 + h_off) * 64 + w];
  }
  __shared__ float red[256];
  red[threadIdx.x] = s; __syncthreads();
  for (int st = 128; st > 0; st >>= 1) {
    if (threadIdx.x < st) red[threadIdx.x] += red[threadIdx.x + st];
    __syncthreads();
  }
  if (threadIdx.x == 0) p[bj] = red[0] * (1.f / 4096.f);
}

// feats[b,i] = sum_j p[b,j]*cw[i,j] + cb[i]
__global__ void feats_kernel(const float* __restrict__ p, const float* __restrict__ cw,
                             const float* __restrict__ cb, float* __restrict__ feats) {
  int t = blockIdx.x * 256 + threadIdx.x;
  if (t >= B_SZ * 16) return;
  int b = t >> 4, i = t & 15;
  float s = cb[i];
  for (int j = 0; j < 32; ++j) s += p[b * 32 + j] * cw[i * 32 + j];
  feats[t] = s;
}

// ---------------- Stage 1a: intra-group CRX chain matrices H ----------------
// Block = (l, m): H = C(local2->3)*C(local1->2)*C(local0->1) on nibble.
// Local qubit t of group m <-> nibble bit (3-t). Writes complex hc and realified wh.
__global__ void build_hc(const float* __restrict__ crx1, const float* __restrict__ crx2,
                         float* __restrict__ hc, float* __restrict__ wh) {
  int l = blockIdx.x >> 2, m = blockIdx.x & 3;
  const float* crx = l ? crx2 : crx1;
  __shared__ float2 M[16][16];
  int t = threadIdx.x, j = t >> 4, col = t & 15;
  M[j][col] = make_float2(j == col ? 1.f : 0.f, 0.f);
  __syncthreads();
  for (int g = 0; g < 3; ++g) {
    float th = crx[4 * m + g] * 0.5f;
    float cth = cosf(th), sth = sinf(th);
    int cb = 1 << (3 - g), tb = 1 << (2 - g);   // ctrl qubit 4m+g, tgt 4m+g+1
    float2 a = M[j][col];
    float2 bp = M[j ^ tb][col];
    __syncthreads();
    if (j & cb)  // new = c*a + (-i s)*bp
      M[j][col] = make_float2(cth * a.x + sth * bp.y, cth * a.y - sth * bp.x);
    __syncthreads();
  }
  float2 v = M[j][col];
  ((float2*)hc)[(size_t)blockIdx.x * 256 + t] = v;
  float* W = wh + (size_t)blockIdx.x * 1024;   // realified 32x32
  W[(2 * j) * 32 + 2 * col]         = v.x;
  W[(2 * j) * 32 + 2 * col + 1]     = -v.y;
  W[(2 * j + 1) * 32 + 2 * col]     = v.y;
  W[(2 * j + 1) * 32 + 2 * col + 1] = v.x;
}

// ---------------- Stage 1b: per-batch fused 1q group matrices ---------------
// Block = (l, b, g). U = kron over nibble bits s of u[qubit 4g+3-s], u = Rot*RX.
// For g==0, K0 = H1 * U (folds the qubit0-3 CRX chain in).
__global__ void build_wg(const float* __restrict__ feats,
                         const float* __restrict__ rot1, const float* __restrict__ rot2,
                         const float* __restrict__ hc, float* __restrict__ wg) {
  int bi = blockIdx.x;                  // l*512 + b*4 + g
  int l = bi >> 9, b = (bi >> 2) & 127, g = bi & 3;
  const float* rot = l ? rot2 : rot1;
  __shared__ float2 u[4][2][2];
  __shared__ float2 U[16][16];
  __shared__ float2 H[16][16];
  int t = threadIdx.x;
  if (t < 4) {
    int q = 4 * g + t;
    float f = feats[b * 16 + q] * 0.5f;
    float cf = cosf(f), sf = sinf(f);
    float2 rx[2][2];
    rx[0][0] = make_float2(cf, 0.f);  rx[0][1] = make_float2(0.f, -sf);
    rx[1][0] = make_float2(0.f, -sf); rx[1][1] = make_float2(cf, 0.f);
    float phi = rot[q * 3 + 0], th = rot[q * 3 + 1] * 0.5f, om = rot[q * 3 + 2];
    float ct = cosf(th), stt = sinf(th);
    float am = -0.5f * (phi + om), ap = 0.5f * (phi + om);
    float ad = 0.5f * (phi - om),  adm = -0.5f * (phi - om);
    float2 rm[2][2];
    rm[0][0] = make_float2(cosf(am) * ct, sinf(am) * ct);
    rm[0][1] = make_float2(-cosf(ad) * stt, -sinf(ad) * stt);
    rm[1][0] = make_float2(cosf(adm) * stt, sinf(adm) * stt);
    rm[1][1] = make_float2(cosf(ap) * ct, sinf(ap) * ct);
    for (int a_ = 0; a_ < 2; ++a_)
      for (int c_ = 0; c_ < 2; ++c_) {
        float re = 0.f, im = 0.f;
        for (int m_ = 0; m_ < 2; ++m_) {
          float2 A = rm[a_][m_], Bv = rx[m_][c_];
          re += A.x * Bv.x - A.y * Bv.y;
          im += A.x * Bv.y + A.y * Bv.x;
        }
        u[t][a_][c_] = make_float2(re, im);
      }
  }
  __syncthreads();
  int j = t >> 4, k = t & 15;
  float re = 1.f, im = 0.f;
  for (int s = 0; s < 4; ++s) {
    float2 f2 = u[3 - s][(j >> s) & 1][(k >> s) & 1];
    float nre = re * f2.x - im * f2.y;
    float nim = re * f2.y + im * f2.x;
    re = nre; im = nim;
  }
  float2 val = make_float2(re, im);
  if (g == 0) {                       // K0 = H1 * U  (uniform branch per block)
    U[j][k] = val;
    H[j][k] = ((const float2*)hc)[(size_t)(l * 4) * 256 + t];
    __syncthreads();
    float rre = 0.f, rim = 0.f;
    for (int r = 0; r < 16; ++r) {
      float2 hh = H[j][r], uu = U[r][k];
      rre += hh.x * uu.x - hh.y * uu.y;
      rim += hh.x * uu.y + hh.y * uu.x;
    }
    val = make_float2(rre, rim);
  }
  float* W = wg + (size_t)bi * 1024;
  W[(2 * j) * 32 + 2 * k]         = val.x;
  W[(2 * j) * 32 + 2 * k + 1]     = -val.y;
  W[(2 * j + 1) * 32 + 2 * k]     = val.y;
  W[(2 * j + 1) * 32 + 2 * k + 1] = val.x;
}

// ---------------- Stage 2: |0...0> init -------------------------------------
__global__ void init_state(float2* __restrict__ st) {
  size_t t = (size_t)blockIdx.x * 256 + threadIdx.x;   // 128*65536 amplitudes
  st[t] = make_float2((t & 65535u) == 0 ? 1.f : 0.f, 0.f);
}

// ---------------- Stage 3a: 16x16 group-matrix pass via WMMA ----------------
// Each wave owns 16 columns (b, r..r+15); gathers their 16 complex amplitudes
// (nibble at bit position `pos`), applies realified 32x32 W with 16
// v_wmma_f32_16x16x4_f32 ops, scatters back in place. wStride=4096 (per-batch
// W) or 0 (shared W). EXEC is all-ones throughout (WMMA requirement).
__global__ void apply_group16(float* __restrict__ stateF, const float* __restrict__ Wbase,
                              int wStride, int pos) {
  int lane = threadIdx.x & 31;
  int wave = blockIdx.x * (blockDim.x >> 5) + (threadIdx.x >> 5);  // 0..32767
  int b = wave >> 8;                        // 256 waves per batch element
  int rbase = (wave & 255) << 4;
  int col = lane & 15, hi = lane >> 4;
  int r = rbase + col;                      // 12 "rest" bits
  int rlo = r & ((1 << pos) - 1);
  int rhi = r >> pos;
  size_t colbase = ((size_t)b << 16) | ((size_t)rhi << (pos + 4)) | (size_t)rlo;
  float* colptr = stateF + 2 * colbase;     // amplitude k=0 of this column
  size_t kstride = (size_t)2 << pos;        // float stride between amplitudes

  const float* Wp = Wbase + (size_t)b * wStride;
  // A fragments: A[M,K] -> vgpr v, K = 4*kc + 2*hi + v, M = 16*i + (lane%16)
  v2f a[2][8];
#pragma unroll
  for (int i = 0; i < 2; ++i)
#pragma unroll
    for (int kc = 0; kc < 8; ++kc)
      a[i][kc] = *(const v2f*)(Wp + (16 * i + col) * 32 + 4 * kc + 2 * hi);
  // B fragments: B[K,N] -> K = 4*kc + 2*hi + v == (re,im) of amplitude 2*kc+hi
  v2f bf[8];
#pragma unroll
  for (int kc = 0; kc < 8; ++kc) {
    int kq = 2 * kc + hi;
    bf[kc] = *(const v2f*)(colptr + (size_t)kq * kstride);
  }
  v8f d[2];
  d[0] = (v8f){0.f, 0.f, 0.f, 0.f, 0.f, 0.f, 0.f, 0.f};
  d[1] = d[0];
#pragma unroll
  for (int kc = 0; kc < 8; ++kc) {
    d[0] = __builtin_amdgcn_wmma_f32_16x16x4_f32(false, a[0][kc], false, bf[kc],
                                                 (short)0, d[0], false, false);
    d[1] = __builtin_amdgcn_wmma_f32_16x16x4_f32(false, a[1][kc], false, bf[kc],
                                                 (short)0, d[1], false, false);
  }
  // D rows M = 16*i + e + 8*hi; rows (2q,2q+1) form complex amplitude j = 8i+4hi+q
#pragma unroll
  for (int i = 0; i < 2; ++i)
#pragma unroll
    for (int q = 0; q < 4; ++q) {
      int jamp = 8 * i + 4 * hi + q;
      v2f val = {d[i][2 * q], d[i][2 * q + 1]};
      *(v2f*)(colptr + (size_t)jamp * kstride) = val;
    }
}

// ---------------- Stage 3b: single cross-boundary CRX pass ------------------
__global__ void apply_crx(float2* __restrict__ st, const float* __restrict__ crx,
                          int idx, int cbit, int tbit) {
  size_t t = (size_t)blockIdx.x * 256 + threadIdx.x;  // 128 * 16384 pairs
  int b = (int)(t >> 14);
  unsigned p = (unsigned)(t & 16383u);
  int lo = cbit < tbit ? cbit : tbit;
  int hb = cbit < tbit ? tbit : cbit;
  unsigned nn = ((p >> lo) << (lo + 1)) | (p & ((1u << lo) - 1));
  nn = ((nn >> hb) << (hb + 1)) | (nn & ((1u << hb) - 1));
  unsigned n0 = nn | (1u << cbit);          // ctrl=1, tgt=0
  unsigned n1 = n0 | (1u << tbit);
  float th = crx[idx] * 0.5f;
  float c = cosf(th), s = sinf(th);
  size_t base = (size_t)b << 16;
  float2 a0 = st[base + n0], a1 = st[base + n1];
  st[base + n0] = make_float2(c * a0.x + s * a1.y, c * a0.y - s * a1.x);
  st[base + n1] = make_float2(s * a0.y + c * a1.x, -s * a0.x + c * a1.y);
}

// ---------------- Stage 4: <Z_i> --------------------------------------------
__global__ void ev_kernel(const float2* __restrict__ st, float* __restrict__ evs) {
  int b = blockIdx.x;
  const float2* s = st + ((size_t)b << 16);
  float acc[16];
  for (int i = 0; i < 16; ++i) acc[i] = 0.f;
  for (int n = threadIdx.x; n < 65536; n += 256) {
    float2 v = s[n];
    float pr = v.x * v.x + v.y * v.y;
#pragma unroll
    for (int i = 0; i < 16; ++i)
      acc[i] += ((n >> (15 - i)) & 1) ? -pr : pr;   // qubit i <-> bit 15-i
  }
  __shared__ float red[256];
  for (int i = 0; i < 16; ++i) {
    red[threadIdx.x] = acc[i]; __syncthreads();
    for (int stp = 128; stp > 0; stp >>= 1) {
      if (threadIdx.x < stp) red[threadIdx.x] += red[threadIdx.x + stp];
      __syncthreads();
    }
    if (threadIdx.x == 0) evs[b * 16 + i] = red[0];
    __syncthreads();
  }
}

// ---------------- Stage 5: FC + log_softmax ---------------------------------
__global__ void head_kernel(const float* __restrict__ evs, const float* __restrict__ fcw,
                            const float* __restrict__ fcb, float* __restrict__ out) {
  int b = blockIdx.x, c = threadIdx.x;
  __shared__ float lg[16];
  __shared__ float mx, ls;
  if (c < 16) {
    float s = fcb[c];
    for (int i = 0; i < 16; ++i) s += evs[b * 16 + i] * fcw[c * 16 + i];
    lg[c] = s;
  }
  __syncthreads();
  if (threadIdx.x == 0) {
    float m = lg[0];
    for (int i = 1; i < 16; ++i) m = fmaxf(m, lg[i]);
    float sum = 0.f;
    for (int i = 0; i < 16; ++i) sum += expf(lg[i] - m);
    mx = m; ls = logf(sum);
  }
  __syncthreads();
  if (c < 16) out[b * 16 + c] = lg[c] - mx - ls;
}

// ---------------------------------------------------------------------------
extern "C" void kernel_launch(void* const* d_in, const int* in_sizes, int n_in,
                              void* d_out, int out_size, void* d_ws, size_t ws_size,
                              hipStream_t stream) {
  const float* x    = (const float*)d_in[0];
  const float* cw   = (const float*)d_in[1];
  const float* cb   = (const float*)d_in[2];
  const float* rot1 = (const float*)d_in[3];
  const float* crx1 = (const float*)d_in[4];
  const float* rot2 = (const float*)d_in[5];
  const float* crx2 = (const float*)d_in[6];
  const float* fcw  = (const float*)d_in[7];
  const float* fcb  = (const float*)d_in[8];
  float* out = (float*)d_out;

  float* ws    = (float*)d_ws;          // needs ~72 MB
  float* state = ws;                    // 16,777,216 f (64 MB)
  float* pbuf  = ws + 16777216;         // 4096 f
  float* feats = pbuf + 4096;           // 2048 f
  float* hc    = feats + 2048;          // 4096 f (2x4 complex 16x16)
  float* wh    = hc + 4096;             // 8192 f (2x4 real 32x32)
  float* wg    = wh + 8192;             // 1,048,576 f (2x128x4 real 32x32)
  float* evs   = wg + 1048576;          // 2048 f

  pool_kernel <<<4096, 256, 0, stream>>>(x, pbuf);
  feats_kernel<<<8,    256, 0, stream>>>(pbuf, cw, cb, feats);
  build_hc    <<<8,    256, 0, stream>>>(crx1, crx2, hc, wh);
  build_wg    <<<1024, 256, 0, stream>>>(feats, rot1, rot2, hc, wg);
  init_state  <<<32768,256, 0, stream>>>((float2*)state);

  for (int l = 0; l < 2; ++l) {
    const float* crx = l ? crx2 : crx1;
    // fused (Rot*RX)^{x4} per group; group0 also carries its internal CRX chain
    for (int g = 0; g < 4; ++g)
      apply_group16<<<4096, 256, 0, stream>>>(
          state, wg + (size_t)(l * 512 + g) * 1024, 4096, 12 - 4 * g);
    // exact bracketing of the CRX ring: C(3,4) H2 C(7,8) H3 C(11,12) H4 C(15,0)
    apply_crx   <<<8192, 256, 0, stream>>>((float2*)state, crx, 3, 12, 11);
    apply_group16<<<4096, 256, 0, stream>>>(state, wh + (size_t)(l * 4 + 1) * 1024, 0, 8);
    apply_crx   <<<8192, 256, 0, stream>>>((float2*)state, crx, 7, 8, 7);
    apply_group16<<<4096, 256, 0, stream>>>(state, wh + (size_t)(l * 4 + 2) * 1024, 0, 4);
    apply_crx   <<<8192, 256, 0, stream>>>((float2*)state, crx, 11, 4, 3);
    apply_group16<<<4096, 256, 0, stream>>>(state, wh + (size_t)(l * 4 + 3) * 1024, 0, 0);
    apply_crx   <<<8192, 256, 0, stream>>>((float2*)state, crx, 15, 0, 15);
  }

  ev_kernel  <<<128, 256, 0, stream>>>((const float2*)state, evs);
  head_kernel<<<128, 32,  0, stream>>>(evs, fcw, fcb, out);
}